// MOTSSegClsRecLoss0226Lightn_32555852103993
// MI455X (gfx1250) — compile-verified
//
#include <hip/hip_runtime.h>
#include <stdint.h>

// ---------------------------------------------------------------------------
// MOTS Seg/Cls/Rec loss for gfx1250 (MI455X).
// B=2, C=10, H=512, W=1024. Output: [loss, focal_loss] (float32).
//
// Pipeline (all batched over P = B*25 (batch,instance) pairs on blockIdx.y):
//   k_prep       : spatial_emb = tanh(pred[0:2])+xym, seed = sigmoid(pred[4])
//   k_batch(+red): res / focal / bg-seed per-batch sums (block partials)
//   k_inst(+red) : per-instance cnt / center / sigma-mean  ->  s = exp(10*sm)
//   k_keys       : per-pixel Lovasz error -> 32-bit sort key (gt in LSB),
//                  plus var/seed partial sums
//   4x (k_hist -> k_scan -> k_scatter): LSD radix sort, 8-bit digits,
//                  wave32 ballot stable ranking, LDS integer-atomic histograms,
//                  tiles staged into LDS with GLOBAL_LOAD_ASYNC_TO_LDS (CDNA5
//                  async copy engine, ASYNCcnt) when the builtins exist.
//   k_bgt/k_bgt_scan/k_contrib/k_fin: prefix-sum of gt over sorted order and
//                  pointwise Lovasz-gradient contributions
//   k_combine    : final scalar combination
// Deterministic: only LDS integer atomics + fixed-order tree reductions.
// Workspace requirement: ~240 MB (keys double-buffer dominates).
// ---------------------------------------------------------------------------

#define Bn      2
#define Cn      10
#define Hn      512
#define Wn      1024
#define HWn     (Hn * Wn)          // 524288
#define NINST   25
#define Pn      (Bn * NINST)       // 50
#define THREADS 256
#define IPT     8
#define TILEn   (THREADS * IPT)    // 2048
#define NBn     (HWn / TILEn)      // 256 blocks per image
#define NBI     64                 // blocks for per-instance stats
#define ITI     (HWn / (NBI * THREADS))   // 32 elems/thread

typedef unsigned int u32;

// ---- CDNA5 async global->LDS staging (probe-guarded) -----------------------
#define ASG __attribute__((address_space(1)))   // prints as __device__ in HIP
#define ASL __attribute__((address_space(3)))   // prints as __shared__ in HIP

typedef int v4i __attribute__((vector_size(4 * sizeof(int))));

#if defined(__has_builtin)
#if __has_builtin(__builtin_amdgcn_global_load_async_to_lds_b128)
#define USE_ASYNC_B128 1
#elif __has_builtin(__builtin_amdgcn_global_load_async_to_lds_b32)
#define USE_ASYNC_B32 1
#endif
#endif
#ifndef USE_ASYNC_B128
#define USE_ASYNC_B128 0
#endif
#ifndef USE_ASYNC_B32
#define USE_ASYNC_B32 0
#endif

// Issue a TILEn-u32 (8 KB) contiguous tile copy global->LDS. 256 threads.
// tile must be 16B-aligned for the B128 variant.
__device__ __forceinline__ void tile_issue(const u32* gsrc, u32* tile, int t) {
#if USE_ASYNC_B128
  // 256 lanes x 16B = 4 KB per issue; two issues cover the 8 KB tile.
  __builtin_amdgcn_global_load_async_to_lds_b128(
      (ASG v4i*)(gsrc + t * 4), (ASL v4i*)(tile + t * 4), 0, 0);
  __builtin_amdgcn_global_load_async_to_lds_b128(
      (ASG v4i*)(gsrc + 1024 + t * 4), (ASL v4i*)(tile + 1024 + t * 4), 0, 0);
#elif USE_ASYNC_B32
#pragma unroll
  for (int j = 0; j < IPT; ++j)
    __builtin_amdgcn_global_load_async_to_lds_b32(
        (ASG int*)(gsrc + j * THREADS + t),
        (ASL int*)(tile + j * THREADS + t), 0, 0);
#else
#pragma unroll
  for (int j = 0; j < IPT; ++j) tile[j * THREADS + t] = gsrc[j * THREADS + t];
#endif
}

// Wave waits for its own async copies; caller must __syncthreads() after.
__device__ __forceinline__ void tile_wait() {
#if USE_ASYNC_B128 || USE_ASYNC_B32
#if defined(__has_builtin) && __has_builtin(__builtin_amdgcn_s_wait_asynccnt)
  __builtin_amdgcn_s_wait_asynccnt(0);
#else
  asm volatile("s_wait_asynccnt 0" ::: "memory");
#endif
#endif
}

__device__ __forceinline__ u32 ballot32(bool v) {
#if defined(__has_builtin)
#if __has_builtin(__builtin_amdgcn_ballot_w32)
  return __builtin_amdgcn_ballot_w32(v);
#else
  return (u32)__ballot(v);
#endif
#else
  return (u32)__ballot(v);
#endif
}

// exclusive scan of 256 u32 across a 256-thread block (Hillis-Steele in LDS)
__device__ __forceinline__ u32 excl_scan256(u32 v, u32* lds, int t) {
  lds[t] = v;
  __syncthreads();
  for (int off = 1; off < THREADS; off <<= 1) {
    u32 x = (t >= off) ? lds[t - off] : 0u;
    __syncthreads();
    lds[t] += x;
    __syncthreads();
  }
  u32 incl = lds[t];
  __syncthreads();
  return incl - v;
}

// deterministic fixed-order block sum of 256 floats; result broadcast
__device__ __forceinline__ float blk_reduce256(float v, float* lds, int t) {
  lds[t] = v;
  __syncthreads();
  for (int s = 128; s > 0; s >>= 1) {
    if (t < s) lds[t] += lds[t + s];
    __syncthreads();
  }
  float r = lds[0];
  __syncthreads();
  return r;
}

// ------------------------------ k_prep -------------------------------------
__global__ void k_prep(const float* __restrict__ pred,
                       const float* __restrict__ xym,
                       float* __restrict__ emb, float* __restrict__ seedm) {
  int b = blockIdx.y, blk = blockIdx.x, t = threadIdx.x;
  const float* pb = pred + (size_t)b * Cn * HWn;
  float* eb = emb + (size_t)b * 2 * HWn;
  float* sb = seedm + (size_t)b * HWn;
#pragma unroll
  for (int j = 0; j < IPT; ++j) {
    int idx = blk * TILEn + j * THREADS + t;
    float x0 = pb[idx];
    float x1 = pb[(size_t)HWn + idx];
    float x4 = pb[(size_t)4 * HWn + idx];
    eb[idx]               = tanhf(x0) + xym[idx];
    eb[(size_t)HWn + idx] = tanhf(x1) + xym[(size_t)HWn + idx];
    sb[idx] = 1.f / (1.f + expf(-x4));
  }
}

// ------------------------------ k_batch ------------------------------------
// partials per (b, blk): [res_num, fg_cnt, focal_sum, valid_cnt, bg_seed2]
__global__ void k_batch(const float* __restrict__ pred,
                        const float* __restrict__ ims,
                        const int* __restrict__ labels,
                        const int* __restrict__ inst,
                        const float* __restrict__ seedm,
                        float* __restrict__ bpart) {
  __shared__ float lds[THREADS];
  int b = blockIdx.y, blk = blockIdx.x, t = threadIdx.x;
  size_t bHW = (size_t)b * HWn;
  const float* pb = pred + (size_t)b * Cn * HWn;
  const float* ib = ims + (size_t)b * 3 * HWn;
  float res = 0.f, fgc = 0.f, fls = 0.f, nv = 0.f, bgs = 0.f;
#pragma unroll
  for (int j = 0; j < IPT; ++j) {
    int idx = blk * TILEn + j * THREADS + t;
    int l = labels[bHW + idx];
    int iv = inst[bHW + idx];
    float fg = (iv > 0) ? 1.f : 0.f;
    float d = 0.f;
#pragma unroll
    for (int c = 0; c < 3; ++c) {
      float df = ib[(size_t)c * HWn + idx] - pb[(size_t)(7 + c) * HWn + idx];
      d += df * df;
    }
    res += d * fg;
    fgc += fg;
    float cm0 = pb[(size_t)5 * HWn + idx];
    float cm1 = pb[(size_t)6 * HWn + idx];
    bool valid = (l < 2);
    int lbl = valid ? l : 0;
    float mx = fmaxf(cm0, cm1);
    float lse = mx + logf(expf(cm0 - mx) + expf(cm1 - mx));
    float logpt = ((lbl == 0) ? cm0 : cm1) - lse;
    float pt = expf(logpt);
    float flv = -(1.f - pt) * (1.f - pt) * logpt;
    if (valid) { fls += flv; nv += 1.f; }
    if (l == 0) { float sd = seedm[bHW + idx]; bgs += sd * sd; }
  }
  float r0 = blk_reduce256(res, lds, t);
  float r1 = blk_reduce256(fgc, lds, t);
  float r2 = blk_reduce256(fls, lds, t);
  float r3 = blk_reduce256(nv, lds, t);
  float r4 = blk_reduce256(bgs, lds, t);
  if (t == 0) {
    float* o = bpart + ((size_t)b * NBn + blk) * 5;
    o[0] = r0; o[1] = r1; o[2] = r2; o[3] = r3; o[4] = r4;
  }
}

__global__ void k_batch_red(const float* __restrict__ bpart,
                            float* __restrict__ bvals) {
  __shared__ float lds[THREADS];
  int b = blockIdx.x, t = threadIdx.x;
  for (int q = 0; q < 5; ++q) {
    float v = bpart[((size_t)b * NBn + t) * 5 + q];
    float r = blk_reduce256(v, lds, t);
    if (t == 0) bvals[b * 5 + q] = r;
  }
}

// ------------------------------ k_inst -------------------------------------
// partials per (p, blk): [cnt, sum_x, sum_y, sum_sig0, sum_sig1]
__global__ void k_inst(const int* __restrict__ inst,
                       const float* __restrict__ xym,
                       const float* __restrict__ pred,
                       float* __restrict__ ipart) {
  __shared__ float lds[THREADS];
  int p = blockIdx.y, blk = blockIdx.x, t = threadIdx.x;
  int b = p / NINST;
  int iid = p % NINST + 1;
  size_t bHW = (size_t)b * HWn;
  const float* sg0 = pred + (size_t)(b * Cn + 2) * HWn;
  const float* sg1 = pred + (size_t)(b * Cn + 3) * HWn;
  float cnt = 0.f, sx = 0.f, sy = 0.f, s0 = 0.f, s1 = 0.f;
  int base = blk * (HWn / NBI);
  for (int j = 0; j < ITI; ++j) {
    int idx = base + j * THREADS + t;
    float m = (inst[bHW + idx] == iid) ? 1.f : 0.f;
    cnt += m;
    sx += m * xym[idx];
    sy += m * xym[(size_t)HWn + idx];
    s0 += m * sg0[idx];
    s1 += m * sg1[idx];
  }
  float r0 = blk_reduce256(cnt, lds, t);
  float r1 = blk_reduce256(sx, lds, t);
  float r2 = blk_reduce256(sy, lds, t);
  float r3 = blk_reduce256(s0, lds, t);
  float r4 = blk_reduce256(s1, lds, t);
  if (t == 0) {
    float* o = ipart + ((size_t)p * NBI + blk) * 5;
    o[0] = r0; o[1] = r1; o[2] = r2; o[3] = r3; o[4] = r4;
  }
}

// stats per p: [cnt, present, safe, c0, c1, sm0, sm1, s0, s1, pad...]
__global__ void k_inst_red(const float* __restrict__ ipart,
                           float* __restrict__ stats) {
  __shared__ float lds[NBI];
  int p = blockIdx.x, t = threadIdx.x;
  float acc[5];
  for (int q = 0; q < 5; ++q) {
    lds[t] = ipart[((size_t)p * NBI + t) * 5 + q];
    __syncthreads();
    for (int s = NBI / 2; s > 0; s >>= 1) {
      if (t < s) lds[t] += lds[t + s];
      __syncthreads();
    }
    acc[q] = lds[0];
    __syncthreads();
  }
  if (t == 0) {
    float cnt = acc[0];
    float present = (cnt > 0.f) ? 1.f : 0.f;
    float safe = fmaxf(cnt, 1.f);
    float* st = stats + (size_t)p * 12;
    float sm0 = acc[3] / safe, sm1 = acc[4] / safe;
    st[0] = cnt; st[1] = present; st[2] = safe;
    st[3] = acc[1] / safe; st[4] = acc[2] / safe;
    st[5] = sm0; st[6] = sm1;
    st[7] = expf(10.f * sm0); st[8] = expf(10.f * sm1);
  }
}

// ------------------------------ k_keys -------------------------------------
// key = ~((float_bits(err) << 1) | gt)  -- ascending sort of key == descending
// by err (err in [0,2] so bits fit); gt rides in the LSB (tie-order invariant).
__global__ void k_keys(const int* __restrict__ inst,
                       const float* __restrict__ emb,
                       const float* __restrict__ pred,
                       const float* __restrict__ seedm,
                       const float* __restrict__ stats,
                       u32* __restrict__ keys, float* __restrict__ vspart) {
  __shared__ float lds[THREADS];
  int p = blockIdx.y, blk = blockIdx.x, t = threadIdx.x;
  int b = p / NINST;
  int iid = p % NINST + 1;
  const float* st = stats + (size_t)p * 12;
  float c0 = st[3], c1 = st[4], sm0 = st[5], sm1 = st[6], s0 = st[7], s1 = st[8];
  size_t bHW = (size_t)b * HWn;
  const float* e0p = emb + (size_t)b * 2 * HWn;
  const float* e1p = e0p + HWn;
  const float* sg0 = pred + (size_t)(b * Cn + 2) * HWn;
  const float* sg1 = pred + (size_t)(b * Cn + 3) * HWn;
  float va = 0.f, sa = 0.f;
#pragma unroll
  for (int j = 0; j < IPT; ++j) {
    int idx = blk * TILEn + j * THREADS + t;
    if (j + 1 < IPT) __builtin_prefetch(&e0p[idx + THREADS], 0, 0);
    int iv = inst[bHW + idx];
    u32 g = (iv == iid) ? 1u : 0u;
    float m = (float)g;
    float d0 = e0p[idx] - c0, d1 = e1p[idx] - c1;
    float d2v = d0 * d0 * s0 + d1 * d1 * s1;
    float dist = expf(-d2v);
    float err = g ? (2.f - 2.f * dist) : (2.f * dist);
    keys[(size_t)p * HWn + idx] = ~((__float_as_uint(err) << 1) | g);
    float v0 = sg0[idx] - sm0, v1 = sg1[idx] - sm1;
    va += m * (v0 * v0 + v1 * v1);
    float sd = seedm[bHW + idx];
    sa += m * (sd - dist) * (sd - dist);
  }
  float r0 = blk_reduce256(va, lds, t);
  float r1 = blk_reduce256(sa, lds, t);
  if (t == 0) {
    float* o = vspart + ((size_t)p * NBn + blk) * 2;
    o[0] = r0; o[1] = r1;
  }
}

// ---------------------------- radix sort -----------------------------------
__global__ void k_hist(const u32* __restrict__ src, u32* __restrict__ hist,
                       int shift) {
  __shared__ u32 cnt[256];
  __shared__ __align__(16) u32 tile[TILEn];
  int p = blockIdx.y, blk = blockIdx.x, t = threadIdx.x;
  const u32* gsrc = src + (size_t)p * HWn + (size_t)blk * TILEn;
  tile_issue(gsrc, tile, t);   // async copy engine (ASYNCcnt) when available
  cnt[t] = 0u;
  tile_wait();
  __syncthreads();
#pragma unroll
  for (int j = 0; j < IPT; ++j)
    atomicAdd(&cnt[(tile[j * THREADS + t] >> shift) & 255u], 1u);  // LDS atomic
  __syncthreads();
  // layout: hist[p][digit*NBn + blk]  (digit-major -> linear scan = bases)
  hist[(size_t)p * (256 * NBn) + (size_t)t * NBn + blk] = cnt[t];
}

__global__ void k_scan(const u32* __restrict__ hist,
                       u32* __restrict__ histScan) {
  __shared__ u32 lds[THREADS];
  int p = blockIdx.x, t = threadIdx.x;
  size_t base = (size_t)p * (256 * NBn) + (size_t)t * 256;
  u32 s = 0u;
  for (int j = 0; j < 256; ++j) s += hist[base + j];
  u32 run = excl_scan256(s, lds, t);
  for (int j = 0; j < 256; ++j) {
    u32 v = hist[base + j];
    histScan[base + j] = run;
    run += v;
  }
}

__global__ void k_scatter(const u32* __restrict__ src, u32* __restrict__ dst,
                          const u32* __restrict__ histScan, int shift) {
  __shared__ u32 gbase[256];
  __shared__ u32 waveCnt[256 * 8];
  __shared__ u32 waveBase[256 * 8];
  __shared__ u32 offCnt[256];
  __shared__ __align__(16) u32 tile[TILEn];
  int p = blockIdx.y, blk = blockIdx.x, t = threadIdx.x;
  size_t pbase = (size_t)p * HWn;
  const u32* gsrc = src + pbase + (size_t)blk * TILEn;
  tile_issue(gsrc, tile, t);   // overlap async copy with base-offset loads
  gbase[t] = histScan[(size_t)p * (256 * NBn) + (size_t)t * NBn + blk];
  offCnt[t] = 0u;
  tile_wait();
  __syncthreads();
  int lane = t & 31, wid = t >> 5;
  u32 lanemask = (1u << lane) - 1u;
#pragma unroll 1
  for (int r = 0; r < IPT; ++r) {
#pragma unroll
    for (int q = 0; q < 8; ++q) waveCnt[t * 8 + q] = 0u;   // t == digit row
    __syncthreads();
    u32 k = tile[r * THREADS + t];
    u32 d = (k >> shift) & 255u;
    // stable wave32 ballot ranking: AND of 8 per-bit match masks
    u32 match = 0xFFFFFFFFu;
#pragma unroll
    for (int bit = 0; bit < 8; ++bit) {
      u32 bal = ballot32(((d >> bit) & 1u) != 0u);
      match &= ((d >> bit) & 1u) ? bal : ~bal;
    }
    u32 rank = __popc(match & lanemask);
    if (rank == 0u) waveCnt[d * 8 + wid] = (u32)__popc(match);
    __syncthreads();
    // per-digit exclusive scan across the 8 waves (thread t owns digit t)
    u32 run = 0u;
#pragma unroll
    for (int w = 0; w < 8; ++w) {
      u32 v = waveCnt[t * 8 + w];
      waveBase[t * 8 + w] = run;
      run += v;
    }
    __syncthreads();
    u32 local = offCnt[d] + waveBase[d * 8 + wid] + rank;
    dst[pbase + gbase[d] + local] = k;
    __syncthreads();
    offCnt[t] += run;
    __syncthreads();
  }
}

// --------------------------- Lovasz finish ---------------------------------
__global__ void k_bgt(const u32* __restrict__ keys, u32* __restrict__ bgt) {
  __shared__ u32 lds[THREADS];
  int p = blockIdx.y, blk = blockIdx.x, t = threadIdx.x;
  size_t base = (size_t)p * HWn + (size_t)blk * TILEn;
  u32 s = 0u;
#pragma unroll
  for (int j = 0; j < IPT; ++j) s += (~keys[base + j * THREADS + t]) & 1u;
  lds[t] = s;
  __syncthreads();
  for (int st = 128; st > 0; st >>= 1) {
    if (t < st) lds[t] += lds[t + st];
    __syncthreads();
  }
  if (t == 0) bgt[p * NBn + blk] = lds[0];
}

__global__ void k_bgt_scan(const u32* __restrict__ bgt,
                           u32* __restrict__ bgts) {
  __shared__ u32 lds[THREADS];
  int p = blockIdx.x, t = threadIdx.x;
  bgts[p * NBn + t] = excl_scan256(bgt[p * NBn + t], lds, t);
}

__global__ void k_contrib(const u32* __restrict__ keys,
                          const u32* __restrict__ bgts,
                          const float* __restrict__ stats,
                          float* __restrict__ bc) {
  __shared__ u32 ulds[THREADS];
  __shared__ float flds[THREADS];
  int p = blockIdx.y, blk = blockIdx.x, t = threadIdx.x;
  u32 carry = bgts[p * NBn + blk];
  float G = stats[(size_t)p * 12];   // total positives
  size_t base = (size_t)p * HWn + (size_t)blk * TILEn + (size_t)t * IPT;
  const uint4* kp = (const uint4*)(keys + base);   // 32B-aligned
  uint4 a = kp[0], bq = kp[1];
  u32 kk[IPT] = {a.x, a.y, a.z, a.w, bq.x, bq.y, bq.z, bq.w};
  u32 tsum = 0u;
#pragma unroll
  for (int j = 0; j < IPT; ++j) tsum += (~kk[j]) & 1u;
  u32 tbase = excl_scan256(tsum, ulds, t);
  u32 c = carry + tbase;   // inclusive-cumsum running value
  float contrib = 0.f;
  int ibase = blk * TILEn + t * IPT;
#pragma unroll
  for (int j = 0; j < IPT; ++j) {
    u32 orig = ~kk[j];
    u32 gj = orig & 1u;
    c += gj;
    float e = __uint_as_float(orig >> 1);   // error >= 0, relu is identity
    int i = ibase + j;
    float ci = (float)c;
    // J(i,c) = 1 - (G-c)/(G + (i+1) - c); denominators always >= 1
    float Ji = 1.f - (G - ci) / (G + (float)(i + 1) - ci);
    float con;
    if (i > 0) {
      float cp = ci - (float)gj;
      float Jp = 1.f - (G - cp) / (G + (float)i - cp);
      con = Ji - Jp;
    } else {
      con = Ji;
    }
    contrib += e * con;
  }
  flds[t] = contrib;
  __syncthreads();
  for (int s = 128; s > 0; s >>= 1) {
    if (t < s) flds[t] += flds[t + s];
    __syncthreads();
  }
  if (t == 0) bc[p * NBn + blk] = flds[0];
}

// instOut per p: [inst_l, var_l, seed_l, present]
__global__ void k_fin(const float* __restrict__ bc,
                      const float* __restrict__ vspart,
                      const float* __restrict__ stats,
                      float* __restrict__ instOut) {
  __shared__ float lds[THREADS];
  int p = blockIdx.x, t = threadIdx.x;
  float il = blk_reduce256(bc[p * NBn + t], lds, t);
  float vv = blk_reduce256(vspart[((size_t)p * NBn + t) * 2 + 0], lds, t);
  float ss = blk_reduce256(vspart[((size_t)p * NBn + t) * 2 + 1], lds, t);
  if (t == 0) {
    const float* st = stats + (size_t)p * 12;
    float* o = instOut + p * 4;
    o[0] = il;
    o[1] = vv / (2.f * st[2]);     // / (N_SIGMA * safe)
    o[2] = 200.f * ss;             // FG_WEIGHT
    o[3] = st[1];                  // present
  }
}

// ------------------------------ k_combine ----------------------------------
__global__ void k_combine(const float* __restrict__ bvals,
                          const float* __restrict__ instOut,
                          const float* __restrict__ seedw,
                          float* __restrict__ out) {
  if (threadIdx.x != 0 || blockIdx.x != 0) return;
  float resn = 0.f, nfg = 0.f;
  for (int b = 0; b < Bn; ++b) { resn += bvals[b * 5 + 0]; nfg += bvals[b * 5 + 1]; }
  float res_loss = (nfg > 0.f) ? resn / (fmaxf(nfg, 1.f) * 3.f) : 0.f;
  float per = 0.f, clst = 0.f;
  for (int b = 0; b < Bn; ++b) {
    float spm = 0.f, si = 0.f, sv = 0.f, ssd = 0.f;
    for (int k = 0; k < NINST; ++k) {
      const float* o = instOut + (b * NINST + k) * 4;
      float pm = o[3];
      spm += pm; si += o[0] * pm; sv += o[1] * pm; ssd += o[2] * pm;
    }
    float so = fmaxf(spm, 1.f);
    float inst_loss = si / so;
    float var_loss = sv / so;
    float seed_loss = (bvals[b * 5 + 4] + ssd) / (float)HWn;
    float nv = bvals[b * 5 + 3];
    float cls = (nv > 0.f) ? bvals[b * 5 + 2] / fmaxf(nv, 1.f) : 0.f;
    float sw = seedw[b];
    per += (1.f * inst_loss + 10.f * var_loss + 1.f * seed_loss) * sw;
    clst += 1.f * cls * sw;
  }
  out[0] = per / (float)Bn + clst / (float)Bn + res_loss * 0.1f;
  out[1] = clst / (float)Bn;
}

// ---------------------------------------------------------------------------
extern "C" void kernel_launch(void* const* d_in, const int* in_sizes, int n_in,
                              void* d_out, int out_size, void* d_ws,
                              size_t ws_size, hipStream_t stream) {
  (void)in_sizes; (void)n_in; (void)out_size; (void)ws_size;
  const float* pred  = (const float*)d_in[0];
  const float* ims   = (const float*)d_in[1];
  const float* xym   = (const float*)d_in[2];
  const float* seedw = (const float*)d_in[3];
  const int*   inst  = (const int*)d_in[4];
  const int*   labels = (const int*)d_in[5];
  float* out = (float*)d_out;

  char* w = (char*)d_ws;
  auto carve = [&](size_t bytes) -> char* {
    char* r = w;
    w += (bytes + 255) & ~(size_t)255;
    return r;
  };
  float* emb      = (float*)carve((size_t)Bn * 2 * HWn * 4);
  float* seedm    = (float*)carve((size_t)Bn * HWn * 4);
  u32*   keysA    = (u32*)carve((size_t)Pn * HWn * 4);
  u32*   keysB    = (u32*)carve((size_t)Pn * HWn * 4);
  u32*   hist     = (u32*)carve((size_t)Pn * 256 * NBn * 4);
  u32*   histScan = (u32*)carve((size_t)Pn * 256 * NBn * 4);
  float* vspart   = (float*)carve((size_t)Pn * NBn * 2 * 4);
  float* ipart    = (float*)carve((size_t)Pn * NBI * 5 * 4);
  float* stats    = (float*)carve((size_t)Pn * 12 * 4);
  u32*   bgt      = (u32*)carve((size_t)Pn * NBn * 4);
  u32*   bgts     = (u32*)carve((size_t)Pn * NBn * 4);
  float* bcon     = (float*)carve((size_t)Pn * NBn * 4);
  float* bpart    = (float*)carve((size_t)Bn * NBn * 5 * 4);
  float* bvals    = (float*)carve((size_t)Bn * 5 * 4);
  float* instOut  = (float*)carve((size_t)Pn * 4 * 4);

  dim3 gB(NBn, Bn), gP(NBn, Pn);

  k_prep<<<gB, THREADS, 0, stream>>>(pred, xym, emb, seedm);
  k_batch<<<gB, THREADS, 0, stream>>>(pred, ims, labels, inst, seedm, bpart);
  k_batch_red<<<dim3(Bn), THREADS, 0, stream>>>(bpart, bvals);
  k_inst<<<dim3(NBI, Pn), THREADS, 0, stream>>>(inst, xym, pred, ipart);
  k_inst_red<<<dim3(Pn), NBI, 0, stream>>>(ipart, stats);
  k_keys<<<gP, THREADS, 0, stream>>>(inst, emb, pred, seedm, stats, keysA,
                                     vspart);
  // 4-pass LSD radix sort (8-bit digits), ping-pong A<->B; ends in keysA
  for (int pass = 0; pass < 4; ++pass) {
    u32* src = (pass & 1) ? keysB : keysA;
    u32* dst = (pass & 1) ? keysA : keysB;
    int shift = pass * 8;
    k_hist<<<gP, THREADS, 0, stream>>>(src, hist, shift);
    k_scan<<<dim3(Pn), THREADS, 0, stream>>>(hist, histScan);
    k_scatter<<<gP, THREADS, 0, stream>>>(src, dst, histScan, shift);
  }
  k_bgt<<<gP, THREADS, 0, stream>>>(keysA, bgt);
  k_bgt_scan<<<dim3(Pn), THREADS, 0, stream>>>(bgt, bgts);
  k_contrib<<<gP, THREADS, 0, stream>>>(keysA, bgts, stats, bcon);
  k_fin<<<dim3(Pn), THREADS, 0, stream>>>(bcon, vspart, stats, instOut);
  k_combine<<<dim3(1), 64, 0, stream>>>(bvals, instOut, seedw, out);
}